// SGGNN_11501922419474
// MI455X (gfx1250) — compile-verified
//
#include <hip/hip_runtime.h>
#include <hip/hip_bf16.h>

typedef __attribute__((ext_vector_type(2))) float v2f;
typedef __attribute__((ext_vector_type(8))) float v8f;

#define DDIM 128

// ---------------------------------------------------------------------------
// init: zero degree counters (2N ints in ws) and agg buffer (first half of out)
// ---------------------------------------------------------------------------
__global__ void init_kernel(int* __restrict__ cnt, float* __restrict__ agg, int N) {
    int total = N * DDIM;
    for (int i = blockIdx.x * blockDim.x + threadIdx.x; i < total;
         i += gridDim.x * blockDim.x) {
        agg[i] = 0.0f;
        if (i < 2 * N) cnt[i] = 0;
    }
}

// ---------------------------------------------------------------------------
// degree counting: out_deg over src, in_deg over dst
// ---------------------------------------------------------------------------
__global__ void count_kernel(const int* __restrict__ src, const int* __restrict__ dst,
                             int* __restrict__ out_cnt, int* __restrict__ in_cnt, int E) {
    int e = blockIdx.x * blockDim.x + threadIdx.x;
    if (e < E) {
        atomicAdd(&out_cnt[src[e]], 1);
        atomicAdd(&in_cnt[dst[e]], 1);
    }
}

__global__ void deg_kernel(const int* __restrict__ out_cnt, const int* __restrict__ in_cnt,
                           float* __restrict__ out_r, float* __restrict__ in_r, int N) {
    int i = blockIdx.x * blockDim.x + threadIdx.x;
    if (i < N) {
        int oc = out_cnt[i]; if (oc < 1) oc = 1;
        int ic = in_cnt[i];  if (ic < 1) ic = 1;
        out_r[i] = rsqrtf((float)oc);
        in_r[i]  = rsqrtf((float)ic);
    }
}

// ---------------------------------------------------------------------------
// GEMM1: h1n[m][n] = (sum_k hidden[m][k]*out_r[m] * W_h[k][n]),  n in [0,128)
// out_r folded into the A fragment (row scaling commutes with the K-sum).
// Block = 256 threads = 8 waves; each wave owns a 16x16 N-tile of a 16-row
// tile. Two independent WMMA chains (even/odd K steps) for matrix-pipe ILP.
// A frag (16x4 f32): lane M = lane&15; half=lane>>4 selects K pair {0,1}/{2,3}.
// C/D (16x16 f32): VGPR j holds M = j + 8*half, N = lane&15.
// ---------------------------------------------------------------------------
__global__ __launch_bounds__(256) void gemm1_kernel(
    const float* __restrict__ hidden,   // [N,128]
    const float* __restrict__ W_h,      // [128,384] row-major
    const float* __restrict__ out_r,    // [N]
    float* __restrict__ h1n,            // [N,128]
    int N)
{
    const int row0 = blockIdx.x * 16;
    const int wave = threadIdx.x >> 5;
    const int lane = threadIdx.x & 31;
    const int half = lane >> 4;
    const int lm   = lane & 15;
    const int n0   = wave * 16;

    int arow = row0 + lm; if (arow >= N) arow = N - 1;   // tail clamp (N%16==0 here)
    const float* Arow = hidden + (size_t)arow * DDIM;
    const float  r    = out_r[arow];

    v8f acc0 = {}, acc1 = {};
    for (int k = 0; k < DDIM; k += 8) {
        const int k0 = k + half * 2;
        const int k1 = k0 + 4;
        v2f a0, a1, b0, b1;
        a0.x = Arow[k0 + 0] * r;
        a0.y = Arow[k0 + 1] * r;
        a1.x = Arow[k1 + 0] * r;
        a1.y = Arow[k1 + 1] * r;
        b0.x = W_h[(size_t)(k0 + 0) * 384 + n0 + lm];
        b0.y = W_h[(size_t)(k0 + 1) * 384 + n0 + lm];
        b1.x = W_h[(size_t)(k1 + 0) * 384 + n0 + lm];
        b1.y = W_h[(size_t)(k1 + 1) * 384 + n0 + lm];
        acc0 = __builtin_amdgcn_wmma_f32_16x16x4_f32(false, a0, false, b0,
                                                     (short)0, acc0, false, false);
        acc1 = __builtin_amdgcn_wmma_f32_16x16x4_f32(false, a1, false, b1,
                                                     (short)0, acc1, false, false);
    }

    float* orow = h1n + (size_t)(row0 + half * 8) * DDIM + n0 + lm;
    if (row0 + 16 <= N) {               // fast path: no per-element guards
#pragma unroll
        for (int j = 0; j < 8; ++j)
            orow[(size_t)j * DDIM] = acc0[j] + acc1[j];
    } else {
#pragma unroll
        for (int j = 0; j < 8; ++j) {
            int m = row0 + j + half * 8;
            if (m < N)
                h1n[(size_t)m * DDIM + n0 + lm] = acc0[j] + acc1[j];
        }
    }
}

// ---------------------------------------------------------------------------
// scatter: agg[dst] += h1n[src]; two edges per 32-lane group, float4 per lane,
// both gathers issued before the atomics for memory-level parallelism.
// h1n (51MB) is L2-resident on MI455X (192MB L2): gathers hit L2, atomics
// resolve at L2.
// ---------------------------------------------------------------------------
__global__ __launch_bounds__(256) void scatter_kernel(
    const float* __restrict__ h1n, const int* __restrict__ src,
    const int* __restrict__ dst, float* __restrict__ agg, int E)
{
    int g    = (int)((blockIdx.x * (unsigned)blockDim.x + threadIdx.x) >> 5);
    int lane = threadIdx.x & 31;
    int e0 = g * 2;
    if (e0 >= E) return;
    int e1 = e0 + 1;
    bool has1 = (e1 < E);

    int s0 = src[e0], d0 = dst[e0];
    int s1 = has1 ? src[e1] : s0;
    int d1 = has1 ? dst[e1] : d0;

    float4 v0 = ((const float4*)(h1n + (size_t)s0 * DDIM))[lane];
    float4 v1 = ((const float4*)(h1n + (size_t)s1 * DDIM))[lane];

    float* r0 = agg + (size_t)d0 * DDIM + lane * 4;
    atomicAdd(r0 + 0, v0.x);
    atomicAdd(r0 + 1, v0.y);
    atomicAdd(r0 + 2, v0.z);
    atomicAdd(r0 + 3, v0.w);
    if (has1) {
        float* r1 = agg + (size_t)d1 * DDIM + lane * 4;
        atomicAdd(r1 + 0, v1.x);
        atomicAdd(r1 + 1, v1.y);
        atomicAdd(r1 + 2, v1.z);
        atomicAdd(r1 + 3, v1.w);
    }
}

// ---------------------------------------------------------------------------
// fused tail: conv = agg * rsqrt(in_deg);
//   f1 = conv @ W_hf[:, :128]    f2 = conv @ W_hf[:, 128:]
//   h2 = hidden @ W_h[:,128:256] h3 = hidden @ W_h[:,256:384]
//   out = h3 + relu(f1+h2)*f2    -> written to both halves of d_out
// agg lives in d_out's first half; barrier separates this block's reads of its
// 16 agg rows from the in-place overwrite (no cross-block row sharing).
// Four independent WMMA chains give matrix-pipe ILP.
// ---------------------------------------------------------------------------
__global__ __launch_bounds__(256) void fused_out_kernel(
    const float* __restrict__ hidden,   // [N,128]
    const float* __restrict__ W_h,      // [128,384]
    const float* __restrict__ W_hf,     // [128,256]
    const float* __restrict__ in_r,     // [N]
    float* out,                         // [2*N*128]; first half holds agg on entry
    int N)
{
    const int row0 = blockIdx.x * 16;
    const int wave = threadIdx.x >> 5;
    const int lane = threadIdx.x & 31;
    const int half = lane >> 4;
    const int lm   = lane & 15;
    const int n0   = wave * 16;

    int arow = row0 + lm; if (arow >= N) arow = N - 1;
    const float* aggrow = out    + (size_t)arow * DDIM;
    const float* hidrow = hidden + (size_t)arow * DDIM;
    const float  rir    = in_r[arow];

    v8f cf1 = {}, cf2 = {}, ch2 = {}, ch3 = {};
    for (int k = 0; k < DDIM; k += 4) {
        const int k0 = k + half * 2;
        v2f aC, aH;
        aC.x = aggrow[k0 + 0] * rir;
        aC.y = aggrow[k0 + 1] * rir;
        aH.x = hidrow[k0 + 0];
        aH.y = hidrow[k0 + 1];

        const size_t bf0 = (size_t)(k0 + 0) * 256 + n0 + lm;
        const size_t bf1 = (size_t)(k0 + 1) * 256 + n0 + lm;
        const size_t bh0 = (size_t)(k0 + 0) * 384 + n0 + lm;
        const size_t bh1 = (size_t)(k0 + 1) * 384 + n0 + lm;

        v2f bF1, bF2, bH2, bH3;
        bF1.x = W_hf[bf0];        bF1.y = W_hf[bf1];
        bF2.x = W_hf[bf0 + 128];  bF2.y = W_hf[bf1 + 128];
        bH2.x = W_h[bh0 + 128];   bH2.y = W_h[bh1 + 128];
        bH3.x = W_h[bh0 + 256];   bH3.y = W_h[bh1 + 256];

        cf1 = __builtin_amdgcn_wmma_f32_16x16x4_f32(false, aC, false, bF1, (short)0, cf1, false, false);
        cf2 = __builtin_amdgcn_wmma_f32_16x16x4_f32(false, aC, false, bF2, (short)0, cf2, false, false);
        ch2 = __builtin_amdgcn_wmma_f32_16x16x4_f32(false, aH, false, bH2, (short)0, ch2, false, false);
        ch3 = __builtin_amdgcn_wmma_f32_16x16x4_f32(false, aH, false, bH3, (short)0, ch3, false, false);
    }

    __syncthreads();   // all waves done reading agg rows before in-place store

    const size_t second = (size_t)N * DDIM;
    if (row0 + 16 <= N) {               // fast path: no per-element guards
        float* orow = out + (size_t)(row0 + half * 8) * DDIM + n0 + lm;
#pragma unroll
        for (int j = 0; j < 8; ++j) {
            float g = cf1[j] + ch2[j];
            g = g > 0.0f ? g : 0.0f;
            float val = ch3[j] + g * cf2[j];
            orow[(size_t)j * DDIM]          = val;   // hidden_new
            orow[second + (size_t)j * DDIM] = val;   // mean over STEP=1 == hidden_new
        }
    } else {
#pragma unroll
        for (int j = 0; j < 8; ++j) {
            int m = row0 + j + half * 8;
            if (m < N) {
                float g = cf1[j] + ch2[j];
                g = g > 0.0f ? g : 0.0f;
                float val = ch3[j] + g * cf2[j];
                size_t idx = (size_t)m * DDIM + n0 + lm;
                out[idx]          = val;
                out[second + idx] = val;
            }
        }
    }
}

// ---------------------------------------------------------------------------
extern "C" void kernel_launch(void* const* d_in, const int* in_sizes, int n_in,
                              void* d_out, int out_size, void* d_ws, size_t ws_size,
                              hipStream_t stream) {
    const float* hidden = (const float*)d_in[0];
    const int*   src    = (const int*)d_in[1];
    const int*   dst    = (const int*)d_in[2];
    const float* W_h    = (const float*)d_in[3];
    const float* W_hf   = (const float*)d_in[4];
    float* out = (float*)d_out;

    const int N = in_sizes[0] / DDIM;
    const int E = in_sizes[1];

    // workspace layout: [out_cnt N][in_cnt N][out_r N][in_r N][h1n N*128]
    int*   out_cnt = (int*)d_ws;
    int*   in_cnt  = out_cnt + N;
    float* out_r   = (float*)(in_cnt + N);
    float* in_r    = out_r + N;
    float* h1n     = in_r + N;
    float* agg     = out;              // first half of d_out doubles as scratch

    const int T = 256;
    const int rowTiles = (N + 15) / 16;

    init_kernel<<<2048, T, 0, stream>>>(out_cnt, agg, N);
    count_kernel<<<(E + T - 1) / T, T, 0, stream>>>(src, dst, out_cnt, in_cnt, E);
    deg_kernel<<<(N + T - 1) / T, T, 0, stream>>>(out_cnt, in_cnt, out_r, in_r, N);
    gemm1_kernel<<<rowTiles, T, 0, stream>>>(hidden, W_h, out_r, h1n, N);
    scatter_kernel<<<(E + 15) / 16, T, 0, stream>>>(h1n, src, dst, agg, E);
    fused_out_kernel<<<rowTiles, T, 0, stream>>>(hidden, W_h, W_hf, in_r, out, N);
}